// InitE_83150566850941
// MI455X (gfx1250) — compile-verified
//
#include <hip/hip_runtime.h>
#include <math.h>

// ---------------------------------------------------------------------------
// Problem constants (from the reference): x [64,28,256,256] f32 -> pool ->
// gelu(p@W1^T+b1) -> @W2^T+b2 -> per-batch QR of 28x6.
// Roofline: pass over x = 448 MiB @ 23.3 TB/s ~= 20us; everything else ~0.
// ---------------------------------------------------------------------------
#define BATCH   64
#define CH      28
#define HW      65536          // 256*256, contiguous per (b,c)
#define SPLIT   32             // blocks per channel in pass 1
#define RANK    6
#define CPAD    32             // LDS row pad for p/h
#define N2      168            // RANK*CH
#define N2PAD   176            // padded to 11 WMMA tiles of 16
#define NCHAN   (BATCH*CH)     // 1792

typedef __attribute__((ext_vector_type(2))) float v2f;
typedef __attribute__((ext_vector_type(4))) float v4f;
typedef __attribute__((ext_vector_type(8))) float v8f;

__device__ __forceinline__ float wave_red32(float v) {
#pragma unroll
    for (int o = 16; o > 0; o >>= 1) v += __shfl_xor(v, o, 32);
    return v;
}

// Unconditional 8B-aligned float2 load (addresses are pre-clamped in-bounds).
__device__ __forceinline__ v2f ld2(const float* __restrict__ p) {
    return *(const v2f*)p;
}

// ---------------------------------------------------------------------------
// Kernel 1: streaming partial reduction. Grid = NCHAN*SPLIT blocks, 256 thr.
// Each block reduces 2048 contiguous floats of one channel with 2 x b128
// non-temporal loads per thread (single-use stream >> 192MB L2).
// Deterministic tree reduction (no float atomics) -> replay-stable.
// ---------------------------------------------------------------------------
__global__ __launch_bounds__(256) void pool_partial_kernel(
    const float* __restrict__ x, float* __restrict__ part) {
    const int blk  = blockIdx.x;
    const int chan = blk / SPLIT;
    const int s    = blk % SPLIT;
    const v4f* xp  = (const v4f*)(x + (size_t)chan * HW + (size_t)s * (HW / SPLIT));
    const int  t   = threadIdx.x;

    v4f a = __builtin_nontemporal_load(xp + t);          // floats [4t .. 4t+3]
    v4f b = __builtin_nontemporal_load(xp + 256 + t);    // floats [1024+4t ..]
    float sum = (a.x + a.y + a.z + a.w) + (b.x + b.y + b.z + b.w);

    sum = wave_red32(sum);
    __shared__ float wsum[8];
    const int wave = t >> 5, lane = t & 31;
    if (lane == 0) wsum[wave] = sum;
    __syncthreads();
    if (t == 0) {
        float tot = 0.f;
#pragma unroll
        for (int i = 0; i < 8; ++i) tot += wsum[i];
        part[blk] = tot;
    }
}

// ---------------------------------------------------------------------------
// Kernel 2: single workgroup (256 thr = 8 waves) does:
//   A) reduce SPLIT partials/channel -> p (mean) in LDS
//   B) h = gelu(p @ W1^T + b1)   via V_WMMA_F32_16X16X4_F32 (4x2 tiles)
//   C) y = h @ W2^T + b2         via V_WMMA_F32_16X16X4_F32 (4x11 tiles)
//   D) per-batch MGS QR of y^T (28x6), one batch per wave iteration
// WMMA layouts per ISA 7.12.2:
//   A 16x4 f32 : lane l holds row M=l&15; VGPR j holds K = 2*(l>>4)+j
//   B 4x16 f32 : lane l holds col N=l&15; VGPR j holds K = 2*(l>>4)+j
//   C/D 16x16  : lane l col N=l&15; VGPR v row M = v + 8*(l>>4)
// Edge columns are handled with clamped (always in-bounds) addresses and a
// 0/1 mask multiply -> unconditional b64 loads + v_cndmask, no EXEC dance
// (WMMA requires EXEC all-ones; control flow here is wave-uniform).
// ---------------------------------------------------------------------------
__global__ __launch_bounds__(256) void head_kernel(
    const float* __restrict__ part,
    const float* __restrict__ W1, const float* __restrict__ b1,
    const float* __restrict__ W2, const float* __restrict__ b2,
    float* __restrict__ out) {
    __shared__ float p_lds[BATCH * CPAD];    // 8 KiB
    __shared__ float h_lds[BATCH * CPAD];    // 8 KiB
    __shared__ float y_lds[BATCH * N2PAD];   // 44 KiB

    const int tid  = threadIdx.x;
    const int lane = tid & 31;
    const int wave = tid >> 5;
    const int half = lane >> 4;   // 0 or 1 (lane group)
    const int l16  = lane & 15;

    // ---- Phase A: partials -> mean, padded to CPAD cols (zeros beyond CH)
    for (int idx = tid; idx < BATCH * CPAD; idx += 256) {
        const int c   = idx & (CPAD - 1);
        const int row = idx >> 5;             // CPAD == 32
        float v = 0.f;
        if (c < CH) {
            const float* pp = part + (row * CH + c) * SPLIT;
            float s = 0.f;
#pragma unroll
            for (int i = 0; i < SPLIT; ++i) s += pp[i];
            v = s * (1.0f / (float)HW);
        }
        p_lds[idx] = v;
    }
    __syncthreads();

    // ---- Phase B: h = gelu(p @ W1^T + b1); 8 waves = 4(M) x 2(N) tiles
    {
        const int   tileM = wave >> 1;        // 0..3
        const int   tileN = wave & 1;         // 0..1
        const int   n     = tileN * 16 + l16;
        const int   nc    = (n < CH) ? n : (CH - 1);   // clamped, in-bounds
        const float wn    = (n < CH) ? 1.f : 0.f;      // edge mask
        v8f acc = {0.f, 0.f, 0.f, 0.f, 0.f, 0.f, 0.f, 0.f};
#pragma unroll
        for (int kk = 0; kk < 7; ++kk) {      // K = 28 = 7*4
            const int k0 = kk * 4 + half * 2; // even -> 8B aligned
            v2f av = ld2(&p_lds[(tileM * 16 + l16) * CPAD + k0]);
            v2f bv = ld2(&W1[nc * CH + k0]);  // B[k][n] = W1[n][k]
            bv.x *= wn; bv.y *= wn;
            acc = __builtin_amdgcn_wmma_f32_16x16x4_f32(
                false, av, false, bv, (short)0, acc, false, false);
        }
        const float bias = (n < CH) ? b1[nc] : 0.f;
#pragma unroll
        for (int v = 0; v < 8; ++v) {
            const int   row = tileM * 16 + v + half * 8;
            const float val = acc[v] + bias;
            const float g   = 0.5f * val * (1.0f + erff(val * 0.70710678118654752f));
            h_lds[row * CPAD + n] = g * wn;   // zero padded cols 28..31
        }
    }
    __syncthreads();

    // ---- Phase C: y = h @ W2^T + b2; 44 tiles (4M x 11N) over 8 waves
    for (int t2 = wave; t2 < 44; t2 += 8) {
        const int   tileM = t2 & 3;
        const int   tileN = t2 >> 2;          // 0..10
        const int   n     = tileN * 16 + l16;
        const int   nc    = (n < N2) ? n : (N2 - 1);
        const float wn    = (n < N2) ? 1.f : 0.f;
        v8f acc = {0.f, 0.f, 0.f, 0.f, 0.f, 0.f, 0.f, 0.f};
#pragma unroll
        for (int kk = 0; kk < 7; ++kk) {
            const int k0 = kk * 4 + half * 2;
            v2f av = ld2(&h_lds[(tileM * 16 + l16) * CPAD + k0]);
            v2f bv = ld2(&W2[nc * CH + k0]);  // B[k][n] = W2[n][k]
            bv.x *= wn; bv.y *= wn;
            acc = __builtin_amdgcn_wmma_f32_16x16x4_f32(
                false, av, false, bv, (short)0, acc, false, false);
        }
        const float bias = (n < N2) ? b2[nc] : 0.f;
#pragma unroll
        for (int v = 0; v < 8; ++v) {
            const int row = tileM * 16 + v + half * 8;
            y_lds[row * N2PAD + n] = acc[v] + bias;
        }
    }
    __syncthreads();

    // ---- Phase D: per-batch QR (modified Gram-Schmidt on 28x6 = y^T),
    // lane = row index (lanes 28..31 carry zeros -> butterfly reduce safe)
    for (int b = wave; b < BATCH; b += 8) {
        float col[RANK];
#pragma unroll
        for (int r = 0; r < RANK; ++r)
            col[r] = (lane < CH) ? y_lds[b * N2PAD + r * CH + lane] : 0.f;
#pragma unroll
        for (int j = 0; j < RANK; ++j) {
#pragma unroll
            for (int i = 0; i < j; ++i) {
                const float d = wave_red32(col[i] * col[j]);
                col[j] -= d * col[i];
            }
            const float nrm = wave_red32(col[j] * col[j]);
            col[j] *= rsqrtf(fmaxf(nrm, 1e-30f));
        }
        if (lane < CH) {
#pragma unroll
            for (int r = 0; r < RANK; ++r)
                out[(b * RANK + r) * CH + lane] = col[r];   // Q^T [B,rank,C]
        }
    }
}

// ---------------------------------------------------------------------------
extern "C" void kernel_launch(void* const* d_in, const int* in_sizes, int n_in,
                              void* d_out, int out_size, void* d_ws, size_t ws_size,
                              hipStream_t stream) {
    const float* x  = (const float*)d_in[0];
    const float* W1 = (const float*)d_in[1];
    const float* b1 = (const float*)d_in[2];
    const float* W2 = (const float*)d_in[3];
    const float* b2 = (const float*)d_in[4];
    float*       out  = (float*)d_out;
    float*       part = (float*)d_ws;   // NCHAN*SPLIT floats = 224 KiB

    pool_partial_kernel<<<NCHAN * SPLIT, 256, 0, stream>>>(x, part);
    head_kernel<<<1, 256, 0, stream>>>(part, W1, b1, W2, b2, out);
}